// GP_EP_RegressionLayer_25778393711190
// MI455X (gfx1250) — compile-verified
//
#include <hip/hip_runtime.h>

typedef __attribute__((ext_vector_type(2))) float v2f;
typedef __attribute__((ext_vector_type(8))) float v8f;

// Problem dimensions (fixed by the reference)
constexpr int kKn = 4;
constexpr int kM  = 128;
constexpr int kD  = 64;
constexpr int kT  = 256;
constexpr int kN  = kKn * kM;      // 512 columns (k,m)
constexpr int kTM = kT - 2;        // 254 middle rows
constexpr int kTP = 256;           // padded to 256 for WMMA tiling
constexpr long long kNTOT = (long long)kKn * kM * kD * kT;  // 8388608

// Workspace layout (floats):
//   SigL : kD * kTP * kTP      (Sigma, overwritten with L, zero-padded)
//   Amu  : kD * kTP * 2        (A_mu, zero-padded rows >= 254)

// ---------------------------------------------------------------------------
// Kernel 1: per-d Sigma (padded 256x256) and A_mu
// ---------------------------------------------------------------------------
__global__ __launch_bounds__(256) void prep_kernel(const float* __restrict__ log_tau,
                                                   float* __restrict__ SigL,
                                                   float* __restrict__ Amu) {
  const int d   = blockIdx.x;
  const int tid = threadIdx.x;
  __shared__ float kp0[kTP], kp1[kTP], am0[kTP], am1[kTP];

  const float tau    = expf(log_tau[d]);
  const float inv2t2 = 1.0f / (2.0f * tau * tau);
  const float bEP    = expf(-(float)((kT - 1) * (kT - 1)) * inv2t2);
  float det = 1.0f - bEP * bEP;
  det = fmaxf(det, 1e-30f);
  const float rdet = 1.0f / det;

  {
    const int t = tid;                    // 256 threads cover 256 padded rows
    float k0 = 0.f, k1 = 0.f, a0 = 0.f, a1 = 0.f;
    if (t < kTM) {
      const float dt0 = (float)(t + 1);       // lag to endpoint 0
      const float dt1 = (float)(kTM - t);     // lag to endpoint T-1
      k0 = expf(-dt0 * dt0 * inv2t2);
      k1 = expf(-dt1 * dt1 * inv2t2);
      a0 = (k0 - bEP * k1) * rdet;            // K_predEP @ K_EPEP_inv
      a1 = (k1 - bEP * k0) * rdet;
    }
    kp0[t] = k0; kp1[t] = k1; am0[t] = a0; am1[t] = a1;
    Amu[((size_t)d * kTP + t) * 2 + 0] = a0;
    Amu[((size_t)d * kTP + t) * 2 + 1] = a1;
  }
  __syncthreads();

  float* S = SigL + (size_t)d * kTP * kTP;
  for (int idx = tid; idx < kTP * kTP; idx += 256) {
    const int t = idx >> 8, s = idx & 255;
    float v = 0.f;
    if (t < kTM && s < kTM) {
      const float dd = (float)(t - s);
      v = expf(-dd * dd * inv2t2) - am0[t] * kp0[s] - am1[t] * kp1[s];
      if (t == s) v += 1e-5f;
    }
    S[idx] = v;
  }
}

// ---------------------------------------------------------------------------
// Kernel 2: in-place Cholesky per d (one block per d), upper-triangle zeroing,
//           and log|det| accumulation into the scalar output.
// ---------------------------------------------------------------------------
__global__ __launch_bounds__(256) void chol_kernel(float* __restrict__ SigL,
                                                   float* __restrict__ out_scalar) {
  const int d   = blockIdx.x;
  const int tid = threadIdx.x;
  float* A = SigL + (size_t)d * kTP * kTP;
  __shared__ float colv[kTP];
  __shared__ float red[256];

  for (int j = 0; j < kTM; ++j) {
    if (tid == 0) {
      float v = A[j * kTP + j];
      v = sqrtf(fmaxf(v, 1e-30f));
      A[j * kTP + j] = v;
      colv[j] = v;
    }
    __syncthreads();
    const float Ljj = colv[j];
    for (int i = j + 1 + tid; i < kTM; i += 256) {
      const float v = A[i * kTP + j] / Ljj;
      A[i * kTP + j] = v;
      colv[i] = v;                           // cache normalized column
    }
    __syncthreads();
    for (int i = j + 1 + tid; i < kTM; i += 256) {
      const float li = colv[i];
      float* rowp = A + i * kTP;
      for (int k = j + 1; k <= i; ++k) rowp[k] -= li * colv[k];
    }
    __syncthreads();
  }

  // Zero the strict upper triangle so WMMA A-tiles can load unconditionally.
  for (int idx = tid; idx < kTP * kTP; idx += 256) {
    const int t = idx >> 8, s = idx & 255;
    if (s > t && t < kTM) A[idx] = 0.f;
  }

  // log|det| = sum log|L_jj|
  red[tid] = (tid < kTM) ? logf(fabsf(A[tid * kTP + tid])) : 0.f;
  __syncthreads();
  for (int off = 128; off > 0; off >>= 1) {
    if (tid < off) red[tid] += red[tid + off];
    __syncthreads();
  }
  if (tid == 0) atomicAdd(out_scalar, red[0]);
}

// ---------------------------------------------------------------------------
// Kernel 3: scalar init + endpoint passthrough
// ---------------------------------------------------------------------------
__global__ void init_scalar_kernel(const float* __restrict__ sldj,
                                   float* __restrict__ out_scalar) {
  if (threadIdx.x == 0 && blockIdx.x == 0) out_scalar[0] = sldj[0];
}

__global__ __launch_bounds__(256) void endpoint_kernel(const float* __restrict__ z,
                                                       float* __restrict__ out) {
  const int idx = blockIdx.x * 256 + threadIdx.x;   // idx = n*D + d
  if (idx < kN * kD) {
    const size_t base = (size_t)idx * kT;
    out[base]          = z[base];
    out[base + kT - 1] = z[base + kT - 1];
  }
}

// ---------------------------------------------------------------------------
// Kernel 4: z_out_middle = A_mu . z_EP  +  L @ z_middle, via f32 WMMA.
// One wave32 per 16x16 output tile. Tiles: d (64) x rowTile (16) x colTile (32).
// K-loop unrolled x4 (two independent accumulators) with a scalar (SGPR) trip
// count via readfirstlane so EXEC is structurally all-ones around every WMMA.
// ---------------------------------------------------------------------------
__global__ __launch_bounds__(256) void apply_kernel(const float* __restrict__ z,
                                                    const float* __restrict__ SigL,
                                                    const float* __restrict__ Amu,
                                                    float* __restrict__ out) {
  const int wave = threadIdx.x >> 5;
  const int lane = threadIdx.x & 31;
  const int tile = blockIdx.x * 8 + wave;          // 0 .. 32767, exact cover
  const int d    = tile >> 9;                       // /512
  const int rem  = tile & 511;
  const int rowBase = (rem >> 5) << 4;              // 0,16,...,240
  const int colBase = (rem & 31) << 4;              // 0,16,...,496

  // Lane decomposition per ISA 16x16x4 F32 WMMA operand maps.
  const int m  = lane & 15;       // A row / B column / C column
  const int hi = lane >> 4;       // half-wave selector
  const int kb = hi << 1;         // K sub-offset {0,2}

  const float* Lp   = SigL + (size_t)d * kTP * kTP + (size_t)(rowBase + m) * kTP;
  const int    n    = colBase + m;                          // output column (k*M+m)
  const float* zcol = z + ((size_t)n * kD + d) * kT;        // contiguous in T

  v8f acc0 = {};
  v8f acc1 = {};
  // Wave-uniform trip count forced into an SGPR -> scalar loop, full EXEC.
  const int smax = __builtin_amdgcn_readfirstlane(rowBase + 16);  // multiple of 16

  for (int s = 0; s < smax; s += 16) {
    // Prefetch next L block for this lane's row (global_prefetch_b8).
    __builtin_prefetch(Lp + s + 16, 0, 3);
#pragma unroll
    for (int u = 0; u < 4; ++u) {
      const int sk = s + 4 * u + kb;
      v2f av, bv;
      // A: L[rowBase+m][sk .. sk+1]  (8B-aligned pair -> global_load_b64)
      av.x = Lp[sk];
      av.y = Lp[sk + 1];
      // B: z_middle[sk .. sk+1][n]  (z_middle[r] = zcol[1+r])
      bv.x = zcol[1 + sk];                                      // 1+sk <= 255
      const int iy = (sk + 2 <= kT - 1) ? (sk + 2) : (kT - 1);  // clamp padded row 255
      bv.y = zcol[iy];                      // L column 255 is zero, value unused
      if (u & 1) {
        acc1 = __builtin_amdgcn_wmma_f32_16x16x4_f32(false, av, false, bv, (short)0,
                                                     acc1, false, false);
      } else {
        acc0 = __builtin_amdgcn_wmma_f32_16x16x4_f32(false, av, false, bv, (short)0,
                                                     acc0, false, false);
      }
    }
  }
  const v8f acc = acc0 + acc1;

  // Add conditioned mean and store middle rows.
  const float zep0 = zcol[0];
  const float zep1 = zcol[kT - 1];
  const float* amu = Amu + (size_t)d * kTP * 2;
  float* ocol = out + ((size_t)n * kD + d) * kT;
#pragma unroll
  for (int v = 0; v < 8; ++v) {
    const int row = rowBase + v + (hi << 3);      // C/D map: VGPR v -> M = v (+8 hi half)
    if (row < kTM) {
      const float mu = amu[row * 2] * zep0 + amu[row * 2 + 1] * zep1;
      ocol[1 + row] = acc[v] + mu;
    }
  }
}

// ---------------------------------------------------------------------------
extern "C" void kernel_launch(void* const* d_in, const int* in_sizes, int n_in,
                              void* d_out, int out_size, void* d_ws, size_t ws_size,
                              hipStream_t stream) {
  const float* z       = (const float*)d_in[0];   // (4,128,64,256) f32
  const float* sldj    = (const float*)d_in[1];   // scalar
  const float* log_tau = (const float*)d_in[2];   // (64,)
  float* out = (float*)d_out;                     // 8388608 z_out + 1 scalar

  float* SigL = (float*)d_ws;                               // 16 MB
  float* Amu  = SigL + (size_t)kD * kTP * kTP;              // 128 KB
  float* out_scalar = out + kNTOT;

  init_scalar_kernel<<<1, 64, 0, stream>>>(sldj, out_scalar);
  prep_kernel<<<kD, 256, 0, stream>>>(log_tau, SigL, Amu);
  chol_kernel<<<kD, 256, 0, stream>>>(SigL, out_scalar);
  endpoint_kernel<<<(kN * kD + 255) / 256, 256, 0, stream>>>(z, out);

  const int tiles = kD * (kTP / 16) * (kN / 16);  // 64*16*32 = 32768
  apply_kernel<<<tiles / 8, 256, 0, stream>>>(z, SigL, Amu, out);
}